// OTDisparityInit_24790551232774
// MI455X (gfx1250) — compile-verified
//
#include <hip/hip_runtime.h>
#include <hip/hip_bf16.h>
#include <cstdint>

typedef __attribute__((ext_vector_type(2))) float v2f;
typedef __attribute__((ext_vector_type(8))) float v8f;

#define NTHREADS 384
#define MAXD     48
#define W_       312            // rows (left pixels)
#define C_       360            // cols (MAXD + W)
#define PITCH    49             // band pitch (floats) -> conflict-free row & diagonal walks
#define NITER    10

__device__ __constant__ const float TAUc   = 0.95f;
__device__ __constant__ const float KAPPAc = 0.025f;               // (1-tau)/2
__device__ __constant__ const float XIc    = 0.0256410256410256f;  // (1-tau)/(1+tau)
__device__ __constant__ const float RHOc   = 19.0f;                // eps*tau/(1-tau)
__device__ __constant__ const float LOGAc  = -5.7430031876770f;    // -log(312)
__device__ __constant__ const float LOGBc  = -5.8861040314502f;    // -log(360)
__device__ __constant__ const float BIGc   = 10000.0f;
#define NEGHUGE (-3.0e38f)

// Block reduction over 384 threads (12 wave32s). is_max selected at compile time.
__device__ __forceinline__ float block_reduce(float v, bool is_max, float ident,
                                              float* red, int tid) {
#pragma unroll
  for (int off = 16; off > 0; off >>= 1) {
    float o = __shfl_xor(v, off, 32);
    v = is_max ? fmaxf(v, o) : (v + o);
  }
  const int wv = tid >> 5, ln = tid & 31;
  __syncthreads();                 // protect red[] reuse across back-to-back reductions
  if (ln == 0) red[wv] = v;
  __syncthreads();
  if (wv == 0) {
    float x = (ln < (NTHREADS / 32)) ? red[ln] : ident;
#pragma unroll
    for (int off = 16; off > 0; off >>= 1) {
      float o = __shfl_xor(x, off, 32);
      x = is_max ? fmaxf(x, o) : (x + o);
    }
    if (ln == 0) red[0] = x;
  }
  __syncthreads();
  float r = red[0];
  __syncthreads();
  return r;
}

__global__ __launch_bounds__(NTHREADS)
void ot_disparity_kernel(const float* __restrict__ scores, float* __restrict__ out) {
  const int tid = threadIdx.x;
  const int bh  = blockIdx.x;          // 0..127 : b = bh/64, h = bh%64
  const int b   = bh >> 6;
  const int h   = bh & 63;

  __shared__ float Ssc[W_ * PITCH];    // scores band, [j][d], pitch 49  (61152 B)
  __shared__ float fS[W_];
  __shared__ float gS[C_];
  __shared__ float red[32];            // total LDS = 63968 B < 64 KB

  // ---- Stage scores slice (d,j) -> LDS transposed band via async global->LDS DMA ----
  {
    const float* base = scores + ((size_t)b * MAXD * 64 + h) * (size_t)W_;
    for (int idx = tid; idx < MAXD * W_; idx += NTHREADS) {
      int d = idx / W_;
      int j = idx - d * W_;
      const float* gp = base + (size_t)d * (64 * W_) + j;          // coalesced in j
      unsigned lds_off = (unsigned)(uintptr_t)(&Ssc[j * PITCH + d]); // per-lane LDS scatter
      asm volatile("global_load_async_to_lds_b32 %0, %1, off"
                   :: "v"(lds_off), "v"(gp) : "memory");
    }
    asm volatile("s_wait_asynccnt 0" ::: "memory");
  }
  __syncthreads();

  // ---- init: f_j = -log_b - lse_d(s[j][d]) ; g never read before first write ----
  if (tid < W_) {
    const int base = tid * PITCH;
    float m = NEGHUGE;
#pragma unroll 8
    for (int d = 0; d < MAXD; ++d) m = fmaxf(m, Ssc[base + d]);
    float s = 0.f;
#pragma unroll 8
    for (int d = 0; d < MAXD; ++d) s += __expf(Ssc[base + d] - m);
    fS[tid] = -LOGBc - (m + __logf(s));
  }
  __syncthreads();

  for (int it = 0; it < NITER; ++it) {
    // (a) reductions over current f: lseF (for empty column 0) and sminF
    float fj   = (tid < W_) ? fS[tid] : NEGHUGE;
    float maxF = block_reduce(fj, true, NEGHUGE, red, tid);
    float sumF = block_reduce((tid < W_) ? __expf(fj - maxF) : 0.f, false, 0.f, red, tid);
    float lseF = maxF + __logf(sumF);
    float x2   = (tid < W_) ? (-fj * (1.0f / RHOc)) : NEGHUGE;
    float max2 = block_reduce(x2, true, NEGHUGE, red, tid);
    float sum2 = block_reduce((tid < W_) ? __expf(x2 - max2) : 0.f, false, 0.f, red, tid);
    float sminF = -RHOc * (LOGAc + max2 + __logf(sum2));

    // (b) g-update: app_c = lse_{j in band(c)} (f_j + s[j][d]),  d = j+48-c
    if (tid < C_) {
      int c = tid;
      float app;
      if (c == 0) {
        app = lseF - BIGc;                         // empty column: exact dense result
      } else {
        int jlo = c - MAXD; if (jlo < 0) jlo = 0;
        int jhi = c;        if (jhi > W_) jhi = W_;
        int idx0 = jlo * (PITCH + 1) + MAXD - c;   // j*49 + (j+48-c), step 50
        float m = NEGHUGE;
        int idx = idx0;
        for (int j = jlo; j < jhi; ++j, idx += PITCH + 1)
          m = fmaxf(m, fS[j] + Ssc[idx]);
        float s = 0.f;
        idx = idx0;
        for (int j = jlo; j < jhi; ++j, idx += PITCH + 1)
          s += __expf(fS[j] + Ssc[idx] - m);
        app = m + __logf(s);
      }
      gS[tid] = TAUc * (LOGBc - app) - KAPPAc * sminF;   // g1
    }
    __syncthreads();

    // (c) sminG1 over g1 ; g = g1 + XI*sminG1 ; sminG = (1+XI)*sminG1 (uniform shift)
    float gv = (tid < C_) ? gS[tid] : 0.f;
    float xg = (tid < C_) ? (-gv * (1.0f / RHOc)) : NEGHUGE;
    float mg = block_reduce(xg, true, NEGHUGE, red, tid);
    float sg = block_reduce((tid < C_) ? __expf(xg - mg) : 0.f, false, 0.f, red, tid);
    float sminG1 = -RHOc * (LOGBc + mg + __logf(sg));
    if (tid < C_) gS[tid] = gv + XIc * sminG1;
    float sminG = (1.0f + XIc) * sminG1;
    __syncthreads();

    // (d) f-update: appF_j = lse_d (g[j+48-d] + s[j][d])
    if (tid < W_) {
      const int j = tid, base = j * PITCH;
      float m = NEGHUGE;
#pragma unroll 8
      for (int d = 0; d < MAXD; ++d) m = fmaxf(m, gS[j + MAXD - d] + Ssc[base + d]);
      float s = 0.f;
#pragma unroll 8
      for (int d = 0; d < MAXD; ++d) s += __expf(gS[j + MAXD - d] + Ssc[base + d] - m);
      fS[j] = TAUc * (LOGAc - (m + __logf(s))) - KAPPAc * sminG;   // f1
    }
    __syncthreads();

    // (e) sminF1 over f1 ; f = f1 + XI*sminF1
    float f1 = (tid < W_) ? fS[tid] : 0.f;
    float xf = (tid < W_) ? (-f1 * (1.0f / RHOc)) : NEGHUGE;
    float mf = block_reduce(xf, true, NEGHUGE, red, tid);
    float sf = block_reduce((tid < W_) ? __expf(xf - mf) : 0.f, false, 0.f, red, tid);
    float sminF1 = -RHOc * (LOGAc + mf + __logf(sf));
    if (tid < W_) fS[tid] = f1 + XIc * sminF1;
    __syncthreads();
  }

  // ---- Final: P = exp(f + g + s); disparity = (Sum d*P)/(max(Sum P,1e-8)) via WMMA ----
  // A fragment (16x4 f32): lane = 16*half + m holds E[m][k0+2*half] , E[m][k0+2*half+1].
  // B fragment (4x16): ones -> row mass ; B[k][n]=d -> d-weighted sum. Same A reused.
  const int wave = tid >> 5;
  const int lane = tid & 31;
  const int half = lane >> 4;
  const int mrow = lane & 15;
  float* outp = out + (size_t)bh * W_;          // (B,1,H,W) flat

  for (int tile = wave; tile < (W_ + 15) / 16; tile += NTHREADS / 32) {
    const int j0  = tile * 16;
    const int row = j0 + mrow;
    const bool rv = (row < W_);
    const float frow = rv ? fS[row] : 0.f;
    v8f accM = {};   // Sum_d P
    v8f accD = {};   // Sum_d d*P
#pragma unroll
    for (int k0 = 0; k0 < MAXD; k0 += 4) {
      const int d0 = k0 + (half << 1);
      const int d1 = d0 + 1;
      float e0 = 0.f, e1 = 0.f;
      if (rv) {
        e0 = __expf(frow + gS[row + MAXD - d0] + Ssc[row * PITCH + d0]);
        e1 = __expf(frow + gS[row + MAXD - d1] + Ssc[row * PITCH + d1]);
      }
      v2f a    = {e0, e1};
      v2f bOne = {1.0f, 1.0f};
      v2f bDv  = {(float)d0, (float)d1};
      accM = __builtin_amdgcn_wmma_f32_16x16x4_f32(false, a, false, bOne,
                                                   (short)0, accM, false, false);
      accD = __builtin_amdgcn_wmma_f32_16x16x4_f32(false, a, false, bDv,
                                                   (short)0, accD, false, false);
    }
    // D layout: vgpr r, lanes 0-15 -> M=r ; lanes 16-31 -> M=r+8 (all N identical here)
    if (mrow == 0) {
#pragma unroll
      for (int r = 0; r < 8; ++r) {
        int orow = j0 + r + (half << 3);
        if (orow < W_) outp[orow] = accD[r] / fmaxf(accM[r], 1e-8f);
      }
    }
  }
}

extern "C" void kernel_launch(void* const* d_in, const int* in_sizes, int n_in,
                              void* d_out, int out_size, void* d_ws, size_t ws_size,
                              hipStream_t stream) {
  (void)in_sizes; (void)n_in; (void)d_ws; (void)ws_size; (void)out_size;
  const float* scores = (const float*)d_in[0];   // (2,48,64,312) f32
  float* out = (float*)d_out;                    // (2,1,64,312) f32
  ot_disparity_kernel<<<dim3(2 * 64), dim3(NTHREADS), 0, stream>>>(scores, out);
}